// DetectionLoss_26371099197483
// MI455X (gfx1250) — compile-verified
//
#include <hip/hip_runtime.h>
#include <math.h>

#define BATCH 512
#define NPRED 1176
#define NTGT  64
#define NCH   9
#define NBITW ((NPRED + 31) / 32)

typedef float v2f __attribute__((ext_vector_type(2)));
typedef float v8f __attribute__((ext_vector_type(8)));

// Reduce 64 values (2 per lane across a wave32) using the matrix unit:
// D = A(16x4, per-lane partials) x B(4x16, ones) + 0  => D[m][*] = rowsum[m].
// Lanes 0-15 hold rows 0..7 in d[0..7], lanes 16-31 hold rows 8..15.
// Sum the 8 D registers, then fold lane<->lane^16. Result in every lane.
__device__ __forceinline__ float wave_sum64_wmma(float a0, float a1) {
    v2f a;  a[0] = a0;  a[1] = a1;
    v2f bm; bm[0] = 1.0f; bm[1] = 1.0f;
    v8f c = {};
    v8f d = __builtin_amdgcn_wmma_f32_16x16x4_f32(false, a, false, bm,
                                                  (short)0, c, false, false);
    float s = d[0] + d[1] + d[2] + d[3] + d[4] + d[5] + d[6] + d[7];
    s += __shfl_xor(s, 16, 32);
    return s;
}

__global__ __launch_bounds__(256) void det_loss_kernel(
    const float* __restrict__ pred,   // (B, N, 9)
    const float* __restrict__ tbox,   // (B, T, 4)
    const int*   __restrict__ tcls,   // (B, T)
    float*       __restrict__ partial) // (B,)
{
    __shared__ float sbx1[NPRED], sby1[NPRED], sbx2[NPRED], sby2[NPRED], sba[NPRED];
    __shared__ float stx1[NTGT], sty1[NTGT], stx2[NTGT], sty2[NTGT], sta[NTGT];
    __shared__ int   stc[NTGT];
    __shared__ int   smatch[NTGT];
    __shared__ unsigned int sbit[NBITW];
    __shared__ float swsum[8];

    const int b   = blockIdx.x;
    const int tid = threadIdx.x;

    // ---- Phase A: stage targets + zero pos-bitmap ----
    if (tid < NTGT) {
        const float* tr = tbox + ((size_t)b * NTGT + tid) * 4;
        float x1 = tr[0], y1 = tr[1], x2 = tr[2], y2 = tr[3];
        stx1[tid] = x1; sty1[tid] = y1; stx2[tid] = x2; sty2[tid] = y2;
        sta[tid]  = (x2 - x1) * (y2 - y1);
        stc[tid]  = tcls[(size_t)b * NTGT + tid];
    }
    if (tid < NBITW) sbit[tid] = 0u;

    // ---- Phase B: decode all N boxes once into LDS (SoA) ----
    for (int n = tid; n < NPRED; n += 256) {
        const float* r = pred + ((size_t)b * NPRED + n) * NCH;
        __builtin_prefetch(r + 256 * NCH, 0, 1);   // global_prefetch_b8
        float p0 = r[0], p1 = r[1], p2 = r[2], p3 = r[3];
        float cx = (p0 * 2.0f - 1.0f) * 736.0f;    // IMG_W/2
        float cy = (p1 * 2.0f - 1.0f) * 416.0f;    // IMG_H/2
        float w  = expf(p2) * 32.0f;               // IMG_W/(IMG_W//PATCH) = 32
        float h  = expf(p3) * 32.0f;               // IMG_H/(IMG_H//PATCH) = 32
        float x1 = cx - w * 0.5f, y1 = cy - h * 0.5f;
        float x2 = cx + w * 0.5f, y2 = cy + h * 0.5f;
        sbx1[n] = x1; sby1[n] = y1; sbx2[n] = x2; sby2[n] = y2;
        sba[n]  = (x2 - x1) * (y2 - y1);
    }
    __syncthreads();

    // ---- Phase C: per-target argmax-IoU over N (one wave owns 8 targets) ----
    const int wv   = tid >> 5;
    const int lane = tid & 31;
    for (int i = 0; i < NTGT / 8; ++i) {
        int t = wv * (NTGT / 8) + i;
        float tx1 = stx1[t], ty1 = sty1[t], tx2 = stx2[t], ty2 = sty2[t];
        float ta  = sta[t];
        unsigned long long best = 0ull;
        for (int n = lane; n < NPRED; n += 32) {
            float ix1 = fmaxf(sbx1[n], tx1);
            float iy1 = fmaxf(sby1[n], ty1);
            float ix2 = fminf(sbx2[n], tx2);
            float iy2 = fminf(sby2[n], ty2);
            float inter = fmaxf(ix2 - ix1, 0.0f) * fmaxf(iy2 - iy1, 0.0f);
            float uni   = sba[n] + ta - inter;
            float iou   = (uni > 0.0f) ? (inter / uni) : 0.0f;
            // iou >= 0 => float bits monotone; inverted index => first-max tie-break
            unsigned long long key =
                ((unsigned long long)__float_as_uint(iou) << 32) |
                (unsigned long long)(0xFFFFFFFFu - (unsigned int)n);
            if (key > best) best = key;
        }
        for (int off = 16; off > 0; off >>= 1) {
            unsigned int hi = __shfl_xor((unsigned int)(best >> 32), off, 32);
            unsigned int lo = __shfl_xor((unsigned int)(best & 0xFFFFFFFFu), off, 32);
            unsigned long long o = ((unsigned long long)hi << 32) | lo;
            if (o > best) best = o;
        }
        if (lane == 0) {
            int m = (int)(0xFFFFFFFFu - (unsigned int)(best & 0xFFFFFFFFu));
            smatch[t] = m;
            atomicOr(&sbit[m >> 5], 1u << (m & 31));  // idempotent
        }
    }
    __syncthreads();

    // ---- Phase D: losses ----
    float acc0 = 0.0f;  // conf BCE partial
    float acc1 = 0.0f;  // 5*box + cls partial
    for (int n = tid; n < NPRED; n += 256) {
        float x = pred[((size_t)b * NPRED + n) * NCH + 4];
        float posn = (float)((sbit[n >> 5] >> (n & 31)) & 1u);
        // logaddexp(0,x) - x*pos, stable form
        acc0 += fmaxf(x, 0.0f) + log1pf(expf(-fabsf(x))) - x * posn;
    }
    if (tid < NTGT) {
        int t = tid;
        int m = smatch[t];
        float bl = 0.0f;
        float pc[4]  = { sbx1[m], sby1[m], sbx2[m], sby2[m] };
        float tg[4]  = { stx1[t], sty1[t], stx2[t], sty2[t] };
        for (int k = 0; k < 4; ++k) {
            float dd = fabsf(pc[k] - tg[k]);
            bl += (dd < 1.0f) ? (0.5f * dd * dd) : (dd - 0.5f);
        }
        const float* r = pred + ((size_t)b * NPRED + m) * NCH + 5;
        float l0 = r[0], l1 = r[1], l2 = r[2], l3 = r[3];
        float mx  = fmaxf(fmaxf(l0, l1), fmaxf(l2, l3));
        float lse = mx + logf(expf(l0 - mx) + expf(l1 - mx) +
                              expf(l2 - mx) + expf(l3 - mx));
        int c = stc[t];
        float lt = (c == 0) ? l0 : ((c == 1) ? l1 : ((c == 2) ? l2 : l3));
        acc1 += 5.0f * bl + (lse - lt);
    }

    // ---- fixed-order reduction: wave via WMMA, block via LDS ----
    float s = wave_sum64_wmma(acc0, acc1);   // all 256 threads: EXEC all-1s
    if (lane == 0) swsum[wv] = s;
    __syncthreads();
    if (tid == 0) {
        float tot = 0.0f;
        for (int i = 0; i < 8; ++i) tot += swsum[i];
        partial[b] = tot;
    }
}

__global__ __launch_bounds__(256) void det_loss_reduce(
    const float* __restrict__ partial, float* __restrict__ out)
{
    __shared__ float swsum[8];
    int tid = threadIdx.x;
    float a0 = partial[tid];
    float a1 = partial[tid + 256];
    float s = wave_sum64_wmma(a0, a1);
    if ((tid & 31) == 0) swsum[tid >> 5] = s;
    __syncthreads();
    if (tid == 0) {
        float tot = 0.0f;
        for (int i = 0; i < 8; ++i) tot += swsum[i];
        out[0] = tot * (1.0f / (float)BATCH);
    }
}

extern "C" void kernel_launch(void* const* d_in, const int* in_sizes, int n_in,
                              void* d_out, int out_size, void* d_ws, size_t ws_size,
                              hipStream_t stream) {
    (void)in_sizes; (void)n_in; (void)out_size; (void)ws_size;
    const float* pred = (const float*)d_in[0];
    const float* tbox = (const float*)d_in[1];
    const int*   tcls = (const int*)d_in[2];
    float* ws = (float*)d_ws;   // BATCH floats of scratch

    det_loss_kernel<<<BATCH, 256, 0, stream>>>(pred, tbox, tcls, ws);
    det_loss_reduce<<<1, 256, 0, stream>>>(ws, (float*)d_out);
}